// MSI_field_1314259992950
// MI455X (gfx1250) — compile-verified
//
#include <hip/hip_runtime.h>
#include <math.h>

#define HH 960
#define WW 1920
#define NL 16      // NLAYERS
#define LTOT 32    // NLAYERS * NSUB
#define BLOCK 256

static __device__ __forceinline__ float sigmoidf(float x) {
    return 1.0f / (1.0f + expf(-x));
}

// radii = cumsum([DMIN, exp(layer_deltas)]); also store reciprocals.
__global__ void radii_precompute(const float* __restrict__ deltas,
                                 float* __restrict__ rad_ws) {
    if (threadIdx.x == 0 && blockIdx.x == 0) {
        float r = 2.0f;  // DMIN
        rad_ws[0]    = r;
        rad_ws[LTOT] = 1.0f / r;
        for (int i = 1; i < LTOT; ++i) {
            r = r + expf(deltas[i - 1]);
            rad_ws[i]        = r;
            rad_ws[LTOT + i] = 1.0f / r;
        }
    }
}

struct Tap {
    int   i00, i01, i10, i11;
    float w00, w01, w10, w11;
    float mask;
};

// Sphere intersection + equirect mapping + bilinear footprint for one layer.
static __device__ __forceinline__ Tap make_tap(
    float r, float invr,
    float a, float b, float rr, float inv2a,
    float rx, float ry, float rz,
    float dx, float dy, float dz)
{
    Tap t;
    float cq   = rr - r * r;
    float disc = b * b - 4.0f * a * cq;
    float sq   = sqrtf(fmaxf(disc, 0.0f));
    float t0   = (-b + sq) * inv2a;
    float t1   = (-b - sq) * inv2a;
    t.mask = (t0 * t1 < 0.0f) ? 1.0f : 0.0f;

    float px = (rx + dx * t0) * invr;
    float py = (ry + dy * t0) * invr;
    float pz = (rz + dz * t0) * invr;

    float gx = pz;                                         // z-coordinate
    float gy = atan2f(py, -px) * 0.3183098861837907f;      // atan2(y,-x)/pi

    float xf = (gx + 1.0f) * (0.5f * (float)(WW - 1));
    float yf = (gy + 1.0f) * (0.5f * (float)(HH - 1));
    float x0f = floorf(xf), y0f = floorf(yf);
    float wx = xf - x0f,   wy = yf - y0f;

    int x0i = (int)x0f, y0i = (int)y0f;
    int x0 = min(max(x0i,     0), WW - 1);
    int x1 = min(max(x0i + 1, 0), WW - 1);
    int y0 = min(max(y0i,     0), HH - 1);
    int y1 = min(max(y0i + 1, 0), HH - 1);

    t.i00 = y0 * WW + x0;
    t.i01 = y0 * WW + x1;
    t.i10 = y1 * WW + x0;
    t.i11 = y1 * WW + x1;
    t.w00 = (1.0f - wx) * (1.0f - wy);
    t.w01 = wx * (1.0f - wy);
    t.w10 = (1.0f - wx) * wy;
    t.w11 = wx * wy;
    return t;
}

static __device__ __forceinline__ float tap4(const float* __restrict__ base, const Tap& t) {
    return base[t.i00] * t.w00 + base[t.i01] * t.w01 +
           base[t.i10] * t.w10 + base[t.i11] * t.w11;
}

__global__ __launch_bounds__(BLOCK) void msi_render(
    const float* __restrict__ origins,
    const float* __restrict__ directions,
    const float* __restrict__ alpha,
    const float* __restrict__ rgb,
    const float* __restrict__ center,
    const float* __restrict__ rad_ws,
    float* __restrict__ out,
    int nrays)
{
    __shared__ float s_rad[LTOT];
    __shared__ float s_inv[LTOT];
    const int tid = threadIdx.x;
    if (tid < LTOT) {
        s_rad[tid] = rad_ws[tid];
        s_inv[tid] = rad_ws[LTOT + tid];
    }
    __syncthreads();

    const int n = blockIdx.x * BLOCK + tid;
    if (n >= nrays) return;

    const float ox = origins[3 * n],    oy = origins[3 * n + 1],    oz = origins[3 * n + 2];
    const float dx = directions[3 * n], dy = directions[3 * n + 1], dz = directions[3 * n + 2];
    const float cx = center[0], cy = center[1], cz = center[2];

    const float rx = ox - cx, ry = oy - cy, rz = oz - cz;
    const float a  = dx * dx + dy * dy + dz * dz;
    const float b  = 2.0f * (dx * rx + dy * ry + dz * rz);
    const float rr = rx * rx + ry * ry + rz * rz;
    const float inv2a = 1.0f / (2.0f * a);

    const long HW = (long)HH * (long)WW;

    float trans = 1.0f;
    float o0 = 0.0f, o1 = 0.0f, o2 = 0.0f;

    // Two layer-pairs per unrolled body => ~40 independent gathers in flight
    // per wave (LOADcnt budget is 63), doubling per-wave latency hiding.
    #pragma unroll 2
    for (int p = 0; p < NL; ++p) {
        const int lA = 2 * p;
        const int lB = 2 * p + 1;

        // Address/weight math for both layers of the pair first (no mem deps).
        Tap tA = make_tap(s_rad[lA], s_inv[lA], a, b, rr, inv2a, rx, ry, rz, dx, dy, dz);
        Tap tB = make_tap(s_rad[lB], s_inv[lB], a, b, rr, inv2a, rx, ry, rz, dx, dy, dz);

        const float* apA = alpha + (long)lA * HW;
        const float* apB = alpha + (long)lB * HW;
        const float* rp  = rgb + (long)p * 3L * HW;

        // Issue all 20 independent gathers; consumers come after, so the
        // scheduler can keep them all in flight (latency hiding).
        float avA = tap4(apA, tA);
        float c0  = tap4(rp,            tA);   // RGB sampled at even layer's coords
        float c1  = tap4(rp + HW,       tA);
        float c2  = tap4(rp + 2L * HW,  tA);
        float avB = tap4(apB, tB);

        float alA = sigmoidf(avA - 5.0f) * tA.mask;
        float alB = sigmoidf(avB - 5.0f) * tB.mask;
        float r0  = sigmoidf(c0);
        float r1  = sigmoidf(c1);
        float r2  = sigmoidf(c2);

        float wA = alA * trans;  trans *= (1.0f - alA);
        float wB = alB * trans;  trans *= (1.0f - alB);
        float wAB = wA + wB;     // both layers in a pair share the same rgb value

        o0 += wAB * r0;
        o1 += wAB * r1;
        o2 += wAB * r2;
    }

    out[3 * n]     = o0;
    out[3 * n + 1] = o1;
    out[3 * n + 2] = o2;
}

extern "C" void kernel_launch(void* const* d_in, const int* in_sizes, int n_in,
                              void* d_out, int out_size, void* d_ws, size_t ws_size,
                              hipStream_t stream) {
    const float* origins    = (const float*)d_in[0];
    const float* directions = (const float*)d_in[1];
    const float* alpha      = (const float*)d_in[2];
    const float* rgb        = (const float*)d_in[3];
    const float* deltas     = (const float*)d_in[4];
    const float* center     = (const float*)d_in[5];
    float* out    = (float*)d_out;
    float* rad_ws = (float*)d_ws;   // 64 floats: radii + reciprocals

    const int nrays = in_sizes[0] / 3;

    radii_precompute<<<1, 32, 0, stream>>>(deltas, rad_ws);

    const int blocks = (nrays + BLOCK - 1) / BLOCK;
    msi_render<<<blocks, BLOCK, 0, stream>>>(origins, directions, alpha, rgb,
                                             center, rad_ws, out, nrays);
}